// DiffPoolNet_78099685310625
// MI455X (gfx1250) — compile-verified
//
#include <hip/hip_runtime.h>
#include <math.h>

typedef __attribute__((ext_vector_type(2))) float v2f;
typedef __attribute__((ext_vector_type(8))) float v8f;

// ---------------- constants ----------------
#define BATCH   256
#define N1      256
#define FEATS   128
#define HID     64
#define K1      64
#define K2      16
#define M1      (BATCH * N1)   // 65536
#define M2      (BATCH * K1)   // 16384
#define M3      (BATCH * K2)   // 4096
#define NEDGES  2097152
#define EPS_BN  1e-5f

static inline int ceilDiv(int a, int b) { return (a + b - 1) / b; }

// ---------------- WMMA helper ----------------
__device__ __forceinline__ v8f wmma4(v2f a, v2f b, v8f c) {
  // D = A(16x4,f32) * B(4x16,f32) + C(16x16,f32)
  return __builtin_amdgcn_wmma_f32_16x16x4_f32(
      /*neg_a=*/false, a, /*neg_b=*/false, b,
      /*c_mod=*/(short)0, c, /*reuse_a=*/false, /*reuse_b=*/false);
}

// ------- batched GEMM: D[b] = op(A[b]) @ B[b] (optional per-row scale) -------
// TA=false: A is [M x K] row-major. TA=true: A stored [K x M], op(A)=A^T.
// Register blocking: each wave computes an (MT*16) x (NT*16) super-tile.
template <bool TA, int MT, int NT>
__global__ void bmm_kernel(const float* __restrict__ A, long aB, int lda,
                           const float* __restrict__ Bm, long bB, int ldb,
                           float* __restrict__ D, long dB, int ldd,
                           int M, int N, int K,
                           const float* __restrict__ rowScale, long sB) {
  const int lane = threadIdx.x & 31;
  const int wid  = threadIdx.x >> 5;
  const int gN   = N / (16 * NT);               // n super-tile count
  const int tiles = (M / (16 * MT)) * gN;
  const int tile = blockIdx.x * (blockDim.x >> 5) + wid;
  if (tile >= tiles) return;                    // wave-uniform: EXEC stays all-ones
  const int b  = blockIdx.y;
  const int m0 = (tile / gN) * (16 * MT);
  const int n0 = (tile % gN) * (16 * NT);
  A  += (long)b * aB;
  Bm += (long)b * bB;
  D  += (long)b * dB;
  const int lm = lane & 15;                     // A row / B col within 16-tile
  const int kh = lane >> 4;                     // K half-select
  v8f acc[MT][NT] = {};
  for (int k0 = 0; k0 < K; k0 += 4) {
    const int ka = k0 + 2 * kh;
    v2f av[MT];
#pragma unroll
    for (int mi = 0; mi < MT; ++mi) {
      const int row = m0 + 16 * mi + lm;
      if (TA) {
        av[mi].x = A[(long)ka * lda + row];
        av[mi].y = A[(long)(ka + 1) * lda + row];
      } else {
        av[mi].x = A[(long)row * lda + ka];
        av[mi].y = A[(long)row * lda + ka + 1];
      }
    }
#pragma unroll
    for (int nj = 0; nj < NT; ++nj) {
      const int col = n0 + 16 * nj + lm;
      v2f bv;
      bv.x = Bm[(long)ka * ldb + col];
      bv.y = Bm[(long)(ka + 1) * ldb + col];
#pragma unroll
      for (int mi = 0; mi < MT; ++mi)
        acc[mi][nj] = wmma4(av[mi], bv, acc[mi][nj]);
    }
  }
  const int rb = kh * 8;
#pragma unroll
  for (int mi = 0; mi < MT; ++mi) {
#pragma unroll
    for (int i = 0; i < 8; ++i) {
      const int row = m0 + 16 * mi + rb + i;
      float sc = rowScale ? rowScale[(long)b * sB + row] : 1.0f;
#pragma unroll
      for (int nj = 0; nj < NT; ++nj)
        D[(long)row * ldd + n0 + 16 * nj + lm] = acc[mi][nj][i] * sc;
    }
  }
}

// -------- fused SAGE combine: D = relu(A1 @ W1^T + A2 @ W2^T + bias) --------
// A1,A2: [M x K] row-major (A2/W2 nullable). W: [N x K] row-major (used as K x N).
template <int MT, int NT>
__global__ void sage_kernel(const float* __restrict__ A1, int lda1,
                            const float* __restrict__ A2, int lda2,
                            const float* __restrict__ W1,
                            const float* __restrict__ W2,
                            const float* __restrict__ bias,
                            float* __restrict__ D, int ldd,
                            int M, int N, int K) {
  const int lane = threadIdx.x & 31;
  const int wid  = threadIdx.x >> 5;
  const int gN   = N / (16 * NT);
  const int tiles = (M / (16 * MT)) * gN;
  const int tile = blockIdx.x * (blockDim.x >> 5) + wid;
  if (tile >= tiles) return;
  const int m0 = (tile / gN) * (16 * MT);
  const int n0 = (tile % gN) * (16 * NT);
  const int lm = lane & 15;
  const int kh = lane >> 4;
  v8f acc[MT][NT] = {};
  for (int k0 = 0; k0 < K; k0 += 4) {
    const int ka = k0 + 2 * kh;
    v2f av[MT];
#pragma unroll
    for (int mi = 0; mi < MT; ++mi) {
      const int row = m0 + 16 * mi + lm;
      av[mi].x = A1[(long)row * lda1 + ka];
      av[mi].y = A1[(long)row * lda1 + ka + 1];
    }
#pragma unroll
    for (int nj = 0; nj < NT; ++nj) {
      const int col = n0 + 16 * nj + lm;
      v2f bv;
      bv.x = W1[(long)col * K + ka];
      bv.y = W1[(long)col * K + ka + 1];
#pragma unroll
      for (int mi = 0; mi < MT; ++mi)
        acc[mi][nj] = wmma4(av[mi], bv, acc[mi][nj]);
    }
  }
  if (A2) {
    for (int k0 = 0; k0 < K; k0 += 4) {
      const int ka = k0 + 2 * kh;
      v2f av[MT];
#pragma unroll
      for (int mi = 0; mi < MT; ++mi) {
        const int row = m0 + 16 * mi + lm;
        av[mi].x = A2[(long)row * lda2 + ka];
        av[mi].y = A2[(long)row * lda2 + ka + 1];
      }
#pragma unroll
      for (int nj = 0; nj < NT; ++nj) {
        const int col = n0 + 16 * nj + lm;
        v2f bv;
        bv.x = W2[(long)col * K + ka];
        bv.y = W2[(long)col * K + ka + 1];
#pragma unroll
        for (int mi = 0; mi < MT; ++mi)
          acc[mi][nj] = wmma4(av[mi], bv, acc[mi][nj]);
      }
    }
  }
  const int rb = kh * 8;
#pragma unroll
  for (int mi = 0; mi < MT; ++mi) {
#pragma unroll
    for (int i = 0; i < 8; ++i) {
      const int row = m0 + 16 * mi + rb + i;
#pragma unroll
      for (int nj = 0; nj < NT; ++nj) {
        const int col = n0 + 16 * nj + lm;
        D[(long)row * ldd + col] = fmaxf(acc[mi][nj][i] + bias[col], 0.0f);
      }
    }
  }
}

// ---------------- small utility kernels ----------------
__global__ void scatter_adj_kernel(const int* __restrict__ e0,
                                   const int* __restrict__ e1,
                                   float* __restrict__ adj, int nE) {
  int i = blockIdx.x * blockDim.x + threadIdx.x;
  if (i < nE) adj[(long)e0[i] * N1 + (e1[i] & (N1 - 1))] = 1.0f;
}

__global__ void invdeg_kernel(const float* __restrict__ adj,
                              float* __restrict__ invdeg, int rows, int W) {
  int r = blockIdx.x * blockDim.x + threadIdx.x;
  if (r < rows) {
    float s = 0.f;
    for (int j = 0; j < W; ++j) s += adj[(long)r * W + j];
    invdeg[r] = 1.0f / fmaxf(s, 1.0f);
  }
}

__global__ void bn_stats_kernel(const float* __restrict__ X, int ld, int M,
                                float* __restrict__ sums, int C, int rpb) {
  const int c  = blockIdx.x;
  const int r0 = blockIdx.y * rpb;
  const int r1 = min(r0 + rpb, M);
  float s = 0.f, s2 = 0.f;
  for (int r = r0 + threadIdx.x; r < r1; r += blockDim.x) {
    float v = X[(long)r * ld + c];
    s += v; s2 += v * v;
  }
  __shared__ float sh0[256], sh1[256];
  sh0[threadIdx.x] = s; sh1[threadIdx.x] = s2;
  __syncthreads();
  for (int off = 128; off > 0; off >>= 1) {
    if (threadIdx.x < off) {
      sh0[threadIdx.x] += sh0[threadIdx.x + off];
      sh1[threadIdx.x] += sh1[threadIdx.x + off];
    }
    __syncthreads();
  }
  if (threadIdx.x == 0) {
    atomicAdd(&sums[c],     sh0[0]);
    atomicAdd(&sums[C + c], sh1[0]);
  }
}

__global__ void bn_apply_kernel(float* __restrict__ X, int ld, int M, int C,
                                const float* __restrict__ sums,
                                const float* __restrict__ gamma,
                                const float* __restrict__ beta, float invM) {
  long idx = (long)blockIdx.x * blockDim.x + threadIdx.x;
  if (idx >= (long)M * C) return;
  int r = (int)(idx / C), c = (int)(idx % C);
  float mu  = sums[c] * invM;
  float var = sums[C + c] * invM - mu * mu;
  float v = X[(long)r * ld + c];
  X[(long)r * ld + c] = (v - mu) * rsqrtf(var + EPS_BN) * gamma[c] + beta[c];
}

__global__ void softmax_kernel(float* __restrict__ X, int rows, int W) {
  int r = blockIdx.x * blockDim.x + threadIdx.x;
  if (r >= rows) return;
  float* p = X + (long)r * W;
  float mx = p[0];
  for (int j = 1; j < W; ++j) mx = fmaxf(mx, p[j]);
  float s = 0.f;
  for (int j = 0; j < W; ++j) { float e = __expf(p[j] - mx); p[j] = e; s += e; }
  float inv = 1.0f / s;
  for (int j = 0; j < W; ++j) p[j] *= inv;
}

__global__ void mean_kernel(const float* __restrict__ X, float* __restrict__ emb,
                            int Bn, int nodes, int C) {
  int idx = blockIdx.x * blockDim.x + threadIdx.x;
  if (idx < Bn * C) {
    int b = idx / C, c = idx % C;
    float s = 0.f;
    for (int n = 0; n < nodes; ++n) s += X[((long)b * nodes + n) * C + c];
    emb[idx] = s / (float)nodes;
  }
}

__global__ void head_kernel(const float* __restrict__ emb,
                            const float* __restrict__ W1, const float* __restrict__ b1,
                            const float* __restrict__ W2, const float* __restrict__ b2,
                            float* __restrict__ out) {
  const int b = blockIdx.x;
  const int t = threadIdx.x;      // 64 threads
  __shared__ float h[64];
  __shared__ float logit[10];
  __shared__ float lse;
  const float* e = emb + (long)b * 192;
  float s = b1[t];
  for (int k = 0; k < 192; ++k) s += e[k] * W1[t * 192 + k];
  h[t] = fmaxf(s, 0.0f);
  __syncthreads();
  if (t < 10) {
    float z = b2[t];
    for (int k = 0; k < 64; ++k) z += h[k] * W2[t * 64 + k];
    logit[t] = z;
  }
  __syncthreads();
  if (t == 0) {
    float mx = logit[0];
    for (int j = 1; j < 10; ++j) mx = fmaxf(mx, logit[j]);
    float se = 0.f;
    for (int j = 0; j < 10; ++j) se += __expf(logit[j] - mx);
    lse = mx + __logf(se);
  }
  __syncthreads();
  if (t < 10) out[(long)b * 10 + t] = logit[t] - lse;
}

// ---------------- host-side launch helpers ----------------
static void bmm(hipStream_t st, bool ta,
                const float* A, long aB, int lda,
                const float* Bm, long bB, int ldb,
                float* D, long dB, int ldd,
                int M, int N, int K, int batch,
                const float* scale, long sB) {
  const int MT = (M % 32 == 0) ? 2 : 1;
  const int NT = (N % 64 == 0) ? 4 : 1;
  const int tiles = (M / (16 * MT)) * (N / (16 * NT));
  dim3 grid(ceilDiv(tiles, 4), batch);
#define BMM_GO(TAv, MTv, NTv) \
  bmm_kernel<TAv, MTv, NTv><<<grid, 128, 0, st>>>(A, aB, lda, Bm, bB, ldb, D, dB, ldd, M, N, K, scale, sB)
  if (ta) {
    if (MT == 2 && NT == 4)      BMM_GO(true, 2, 4);
    else if (MT == 2)            BMM_GO(true, 2, 1);
    else if (NT == 4)            BMM_GO(true, 1, 4);
    else                         BMM_GO(true, 1, 1);
  } else {
    if (MT == 2 && NT == 4)      BMM_GO(false, 2, 4);
    else if (MT == 2)            BMM_GO(false, 2, 1);
    else if (NT == 4)            BMM_GO(false, 1, 4);
    else                         BMM_GO(false, 1, 1);
  }
#undef BMM_GO
}

static void sage(hipStream_t st, const float* A1, int lda1,
                 const float* A2, int lda2,
                 const float* W1, const float* W2, const float* bias,
                 float* D, int ldd, int M, int N, int K) {
  const int NT = (N % 64 == 0) ? 4 : 1;   // M is always a multiple of 32 here
  const int tiles = (M / 32) * (N / (16 * NT));
  dim3 grid(ceilDiv(tiles, 4));
  if (NT == 4)
    sage_kernel<2, 4><<<grid, 128, 0, st>>>(A1, lda1, A2, lda2, W1, W2, bias, D, ldd, M, N, K);
  else
    sage_kernel<2, 1><<<grid, 128, 0, st>>>(A1, lda1, A2, lda2, W1, W2, bias, D, ldd, M, N, K);
}

static void bn(hipStream_t st, float* X, int ld, int M, int C,
               const float* g, const float* b, float* sums) {
  hipMemsetAsync(sums, 0, 2 * C * sizeof(float), st);
  const int rpb = 4096;
  bn_stats_kernel<<<dim3(C, ceilDiv(M, rpb)), 256, 0, st>>>(X, ld, M, sums, C, rpb);
  long total = (long)M * C;
  bn_apply_kernel<<<ceilDiv((int)total, 256), 256, 0, st>>>(X, ld, M, C, sums, g, b, 1.0f / (float)M);
}

struct ConvP { const float *Wr, *Wroot, *br, *g, *b; int oc; };

static void run_gnn(hipStream_t st, const float* Xin, int ldx, int inC,
                    const float* adj, int nodes, int batch, const float* invdeg,
                    float* cat, int catW, const ConvP cp[3],
                    float* agg, float* bnsums) {
  const float* cur = Xin;
  int ldc = ldx, C = inC, col = 0;
  const int M = batch * nodes;
  for (int i = 0; i < 3; ++i) {
    const int oc = cp[i].oc;
    // agg = (adj @ cur) * invdeg  (mean aggregation)
    bmm(st, false, adj, (long)nodes * nodes, nodes,
        cur, (long)nodes * ldc, ldc,
        agg, (long)nodes * C, C,
        nodes, C, nodes, batch, invdeg, nodes);
    // out = relu(agg @ Wr^T + cur @ Wroot^T + br)
    sage(st, agg, C, cur, ldc, cp[i].Wr, cp[i].Wroot, cp[i].br,
         cat + col, catW, M, oc, C);
    bn(st, cat + col, catW, M, oc, cp[i].g, cp[i].b, bnsums);
    cur = cat + col; ldc = catW; C = oc; col += oc;
  }
}

// ---------------- entry point ----------------
extern "C" void kernel_launch(void* const* d_in, const int* in_sizes, int n_in,
                              void* d_out, int out_size, void* d_ws, size_t ws_size,
                              hipStream_t stream) {
  (void)in_sizes; (void)n_in; (void)out_size; (void)ws_size;
  const float* x    = (const float*)d_in[0];  // [65536, 128]
  const int*   eidx = (const int*)  d_in[2];  // [2, 2097152]
  auto P = [&](int i) { return (const float*)d_in[i]; };

  // workspace layout (floats)
  float* W = (float*)d_ws;
  size_t o = 0;
  auto alloc = [&](size_t n) { float* p = W + o; o += n; return p; };
  float* adj1   = alloc((size_t)M1 * N1);        // 16.7M  [B,256,256]
  float* invdeg = alloc(M1);                     // reused at every level
  float* C1s    = alloc((size_t)M1 * 192);
  float* C1h    = alloc((size_t)M1 * 192);
  float* agg    = alloc((size_t)M1 * 192);       // conv scratch (max width 192)
  float* s1     = alloc((size_t)M1 * K1);
  float* tmpAS  = alloc((size_t)M1 * K1);        // adj @ s
  float* h2     = alloc((size_t)M2 * 192);
  float* adj2   = alloc((size_t)M2 * K1);        // [B,64,64]
  float* bnsum  = alloc(512);
  // level-2/3 buffers alias the adj1 region (free after diff_pool 1)
  float* C2s    = adj1;
  float* C2h    = C2s + (size_t)M2 * 144;
  float* s2     = C2h + (size_t)M2 * 192;
  float* tmpAS2 = s2 + (size_t)M2 * K2;
  float* h3     = tmpAS2 + (size_t)M2 * K2;
  float* adj3   = h3 + (size_t)M3 * 192;         // [B,16,16]
  float* C3     = adj3 + (size_t)M3 * K2;
  float* emb    = C3 + (size_t)M3 * 192;

  // ---- build dense adjacency + inverse degrees ----
  hipMemsetAsync(adj1, 0, (size_t)M1 * N1 * sizeof(float), stream);
  scatter_adj_kernel<<<ceilDiv(NEDGES, 256), 256, 0, stream>>>(eidx, eidx + NEDGES, adj1, NEDGES);
  invdeg_kernel<<<ceilDiv(M1, 256), 256, 0, stream>>>(adj1, invdeg, M1, N1);

  // ---- level 1 ----
  ConvP g1p[3] = {{P(24), P(25), P(26), P(19), P(18), HID},
                  {P(27), P(28), P(29), P(21), P(20), HID},
                  {P(30), P(31), P(32), P(23), P(22), K1}};
  run_gnn(stream, x, FEATS, FEATS, adj1, N1, BATCH, invdeg, C1s, 192, g1p, agg, bnsum);
  sage(stream, C1s, 192, nullptr, 0, P(33), nullptr, P(34), s1, K1, M1, K1, 192);

  ConvP g1e[3] = {{P(9),  P(10), P(11), P(4), P(3), HID},
                  {P(12), P(13), P(14), P(6), P(5), HID},
                  {P(15), P(16), P(17), P(8), P(7), HID}};
  run_gnn(stream, x, FEATS, FEATS, adj1, N1, BATCH, invdeg, C1h, 192, g1e, agg, bnsum);

  // diff_pool 1: s = softmax(s); h2 = s^T h; adj2 = s^T (adj1 s)
  softmax_kernel<<<ceilDiv(M1, 256), 256, 0, stream>>>(s1, M1, K1);
  bmm(stream, true,  s1, (long)N1 * K1, K1, C1h, (long)N1 * 192, 192,
      h2, (long)K1 * 192, 192, K1, 192, N1, BATCH, nullptr, 0);
  bmm(stream, false, adj1, (long)N1 * N1, N1, s1, (long)N1 * K1, K1,
      tmpAS, (long)N1 * K1, K1, N1, K1, N1, BATCH, nullptr, 0);
  bmm(stream, true,  s1, (long)N1 * K1, K1, tmpAS, (long)N1 * K1, K1,
      adj2, (long)K1 * K1, K1, K1, K1, N1, BATCH, nullptr, 0);
  invdeg_kernel<<<ceilDiv(M2, 256), 256, 0, stream>>>(adj2, invdeg, M2, K1);

  // ---- level 2 ----
  ConvP g2p[3] = {{P(56), P(57), P(58), P(51), P(50), HID},
                  {P(59), P(60), P(61), P(53), P(52), HID},
                  {P(62), P(63), P(64), P(55), P(54), K2}};
  run_gnn(stream, h2, 192, 192, adj2, K1, BATCH, invdeg, C2s, 144, g2p, agg, bnsum);
  sage(stream, C2s, 144, nullptr, 0, P(65), nullptr, P(66), s2, K2, M2, K2, 144);

  ConvP g2e[3] = {{P(41), P(42), P(43), P(36), P(35), HID},
                  {P(44), P(45), P(46), P(38), P(37), HID},
                  {P(47), P(48), P(49), P(40), P(39), HID}};
  run_gnn(stream, h2, 192, 192, adj2, K1, BATCH, invdeg, C2h, 192, g2e, agg, bnsum);

  // diff_pool 2
  softmax_kernel<<<ceilDiv(M2, 256), 256, 0, stream>>>(s2, M2, K2);
  bmm(stream, true,  s2, (long)K1 * K2, K2, C2h, (long)K1 * 192, 192,
      h3, (long)K2 * 192, 192, K2, 192, K1, BATCH, nullptr, 0);
  bmm(stream, false, adj2, (long)K1 * K1, K1, s2, (long)K1 * K2, K2,
      tmpAS2, (long)K1 * K2, K2, K1, K2, K1, BATCH, nullptr, 0);
  bmm(stream, true,  s2, (long)K1 * K2, K2, tmpAS2, (long)K1 * K2, K2,
      adj3, (long)K2 * K2, K2, K2, K2, K1, BATCH, nullptr, 0);
  invdeg_kernel<<<ceilDiv(M3, 256), 256, 0, stream>>>(adj3, invdeg, M3, K2);

  // ---- level 3 ----
  ConvP g3e[3] = {{P(73), P(74), P(75), P(68), P(67), HID},
                  {P(76), P(77), P(78), P(70), P(69), HID},
                  {P(79), P(80), P(81), P(72), P(71), HID}};
  run_gnn(stream, h3, 192, 192, adj3, K2, BATCH, invdeg, C3, 192, g3e, agg, bnsum);

  // ---- head: mean over nodes, lin1+relu, lin2, log_softmax ----
  mean_kernel<<<ceilDiv(BATCH * 192, 256), 256, 0, stream>>>(C3, emb, BATCH, K2, 192);
  head_kernel<<<BATCH, 64, 0, stream>>>(emb, P(82), P(83), P(84), P(85), (float*)d_out);
}